// CML2DWithStats_21380347199586
// MI455X (gfx1250) — compile-verified
//
#include <hip/hip_runtime.h>
#include <stdint.h>

// ---------------------------------------------------------------------------
// CML2DWithStats on gfx1250 (MI455X).
// Temporal-blocked stencil: one 64x64 output tile per workgroup, halo = 15
// (one per step), all 15 steps run entirely in LDS. Tile (+halo) is brought
// in with the CDNA5 Tensor Data Mover (tensor_load_to_lds), synchronized via
// s_wait_tensorcnt. Outputs written once, coalesced.
// ---------------------------------------------------------------------------

typedef __attribute__((ext_vector_type(4))) unsigned int v4u_t;
typedef __attribute__((ext_vector_type(8))) int          v8i_t;
typedef __attribute__((ext_vector_type(4))) int          v4i_t;

#define TILE    64
#define HALO    15
#define REG     94          // TILE + 2*HALO
#define STRIDE  96          // LDS row stride (floats)
#define NSTEPS  15

#define R_PARAM   3.9f
#define EPS       0.3f
#define BETA      0.15f
#define CLAMP_LO  1.0e-4f
#define CLAMP_HI  (1.0f - 1.0e-4f)

// Issue one TDM 2D tile load: h rows of w f32 elements, global row stride
// `gstride` elements, into LDS at byte offset lds_byte with row stride
// `lstride` elements. Uses iterate mode: tile = (w x 1), iterated h times.
// D# layout per CDNA5 ISA ch.8 (CDNA5 ISA).
__device__ __forceinline__ void tdm_load_2d(uint32_t lds_byte, uint64_t gaddr,
                                            int w, int h, int gstride, int lstride) {
  const uint32_t TD = 0x100000u;  // large tensor dims -> OOB never triggers
  v4u_t g0;
  g0.x = 1u;                                            // count=1, user D#
  g0.y = lds_byte;                                      // lds_addr (bytes)
  g0.z = (uint32_t)(gaddr & 0xFFFFFFFFull);             // global_addr[31:0]
  g0.w = (uint32_t)((gaddr >> 32) & 0x1FFFFFFull)       // global_addr[56:32]
       | (2u << 30);                                    // type = 2 ("image")
  v8i_t g1;
  g1.s0 = (int)((2u << 16) | (1u << 19));               // data_size=4B, iterate_enable=1
  g1.s1 = (int)((TD & 0xFFFFu) << 16);                  // tensor_dim0[15:0]
  g1.s2 = (int)(((TD >> 16) & 0xFFFFu)                  // tensor_dim0[31:16]
       |  ((TD & 0xFFFFu) << 16));                      // tensor_dim1[15:0]
  g1.s3 = (int)(((TD >> 16) & 0xFFFFu)                  // tensor_dim1[31:16]
       |  ((uint32_t)w << 16));                         // tile_dim0 = w
  g1.s4 = 1;                                            // tile_dim1=1, tile_dim2=0
  g1.s5 = gstride;                                      // tensor_dim0_stride[31:0]
  g1.s6 = 0;                                            // stride hi / dim1_stride lo
  g1.s7 = 0;
  v4i_t g2;
  g2.x = 0;                                             // tensor_dim2 (unused, iterate)
  g2.y = lstride;                                       // lds_addr_increment (elements)
  g2.z = gstride;                                       // global_addr_increment[31:0]
  g2.w = (int)(((uint32_t)(h - 1)) << 16);              // iterate_count | ga_inc[47:32]
  v4i_t g3 = {0, 0, 0, 0};
#if __clang_major__ >= 23
  v8i_t g4 = {0, 0, 0, 0, 0, 0, 0, 0};
  __builtin_amdgcn_tensor_load_to_lds(g0, g1, g2, g3, g4, 0);
#else
  __builtin_amdgcn_tensor_load_to_lds(g0, g1, g2, g3, 0);
#endif
}

__global__ __launch_bounds__(256, 1)
void cml2d_stats_kernel(const float* __restrict__ drive,
                        const float* __restrict__ K_local,
                        float* __restrict__ out, size_t N) {
  __shared__ float G[REG * STRIDE];   // grid state
  __shared__ float M[REG * STRIDE];   // mapped = R*g*(1-g)
  __shared__ float D[REG * STRIDE];   // drive (cached)

  const int tx = threadIdx.x;               // 0..63
  const int ty = threadIdx.y;               // 0..3
  const int tid = ty * 64 + tx;
  const int tileX = blockIdx.x;             // 0..3
  const int tileY = blockIdx.y;             // 0..3
  const int im    = blockIdx.z;             // 0..127  (b*8 + c)
  const int ch    = im & 7;

  const int gx0 = tileX * TILE - HALO;      // region origin (may be negative)
  const int gy0 = tileY * TILE - HALO;

  // per-channel 3x3 taps (uniform -> scalar loads)
  const float* Kc = K_local + ch * 9;
  const float k00 = Kc[0], k01 = Kc[1], k02 = Kc[2];
  const float k10 = Kc[3], k11 = Kc[4], k12 = Kc[5];
  const float k20 = Kc[6], k21 = Kc[7], k22 = Kc[8];

  // Zero-fill G and D: out-of-image halo must stay 0 (== conv zero padding,
  // since mapped(0) = 0 and those cells are never updated).
  for (int i = tid; i < REG * STRIDE; i += 256) { G[i] = 0.0f; D[i] = 0.0f; }
  __syncthreads();

  // Clamp halo'd region to the image and DMA it into LDS with the TDM.
  const int cy0 = max(gy0, 0), cy1 = min(gy0 + REG, 256);
  const int cx0 = max(gx0, 0), cx1 = min(gx0 + REG, 256);
  const int w = cx1 - cx0, h = cy1 - cy0;
  const uint64_t gaddr =
      (uint64_t)(uintptr_t)(drive + (size_t)im * 65536 + (size_t)cy0 * 256 + cx0);
  const uint32_t loff = (uint32_t)(((cy0 - gy0) * STRIDE + (cx0 - gx0)) * 4);

  if (tid < 32) {  // wave 0 issues the DMA (EXEC ignored by TDM; once per wave)
    const uint32_t ldsG = (uint32_t)(uintptr_t)&G[0] + loff;  // flat[31:0] == LDS offset
    const uint32_t ldsD = (uint32_t)(uintptr_t)&D[0] + loff;
    tdm_load_2d(ldsG, gaddr, w, h, 256, STRIDE);
    tdm_load_2d(ldsD, gaddr, w, h, 256, STRIDE);
    __builtin_amdgcn_s_wait_tensorcnt(0);
  }
  __syncthreads();

  float sum[16], sq[16];
#pragma unroll
  for (int j = 0; j < 16; ++j) { sum[j] = 0.0f; sq[j] = 0.0f; }

  for (int s = 0; s < NSTEPS; ++s) {
    // ---- mapped = R*g*(1-g) over region [s, REG-s)^2 ----
    const int lo = s, hi = REG - s;
    for (int y = lo + ty; y < hi; y += 4)
      for (int x = lo + tx; x < hi; x += 64) {
        const float g = G[y * STRIDE + x];
        M[y * STRIDE + x] = R_PARAM * g * (1.0f - g);
      }
    __syncthreads();

    // ---- conv + blend + clamp over region [s+1, REG-s-1)^2 ----
    const int lo2 = s + 1, hi2 = REG - s - 1;
    for (int y = lo2 + ty; y < hi2; y += 4)
      for (int x = lo2 + tx; x < hi2; x += 64) {
        const float* m0 = &M[(y - 1) * STRIDE + x];
        const float* m1 = m0 + STRIDE;
        const float* m2 = m1 + STRIDE;
        float local = k00 * m0[-1];
        local = fmaf(k01, m0[0], local);
        local = fmaf(k02, m0[1], local);
        local = fmaf(k10, m1[-1], local);
        local = fmaf(k11, m1[0], local);
        local = fmaf(k12, m1[1], local);
        local = fmaf(k20, m2[-1], local);
        local = fmaf(k21, m2[0], local);
        local = fmaf(k22, m2[1], local);
        const float mapped = m1[0];
        const float phys = fmaf(EPS, local, (1.0f - EPS) * mapped);
        float g = fmaf(BETA, D[y * STRIDE + x], (1.0f - BETA) * phys);
        g = fminf(fmaxf(g, CLAMP_LO), CLAMP_HI);
        const bool inimg =
            ((unsigned)(gy0 + y) < 256u) && ((unsigned)(gx0 + x) < 256u);
        G[y * STRIDE + x] = inimg ? g : 0.0f;
      }
    __syncthreads();

    // ---- accumulate stats on the 64x64 core (reads only; next write to G
    //      happens after the next barrier, so no race) ----
#pragma unroll
    for (int j = 0; j < 16; ++j) {
      const float g = G[(HALO + ty + 4 * j) * STRIDE + (HALO + tx)];
      sum[j] += g;
      sq[j] = fmaf(g, g, sq[j]);
    }
  }

  // ---- write 5 outputs: last, mean, var, delta, last_drive ----
  float* olast = out;
  float* omean = out + N;
  float* ovar  = out + 2 * N;
  float* odel  = out + 3 * N;
  float* old2  = out + 4 * N;
  const float inv = 1.0f / (float)NSTEPS;
#pragma unroll
  for (int j = 0; j < 16; ++j) {
    const int ly = HALO + ty + 4 * j, lx = HALO + tx;
    const int gy = gy0 + ly, gx = gx0 + lx;  // == tile origin + core coords
    const size_t o = (size_t)im * 65536 + (size_t)gy * 256 + (size_t)gx;
    const float last = G[ly * STRIDE + lx];
    const float dv   = D[ly * STRIDE + lx];
    const float mean = sum[j] * inv;
    const float var  = fmaf(-mean, mean, sq[j] * inv);  // E[g^2] - mean^2
    const float dd   = last - dv;
    olast[o] = last;
    omean[o] = mean;
    ovar[o]  = var;
    odel[o]  = dd;
    old2[o]  = dd;
  }
}

extern "C" void kernel_launch(void* const* d_in, const int* in_sizes, int n_in,
                              void* d_out, int out_size, void* d_ws, size_t ws_size,
                              hipStream_t stream) {
  const float* drive   = (const float*)d_in[0];
  const float* K_local = (const float*)d_in[1];
  float* out = (float*)d_out;
  const size_t N = (size_t)in_sizes[0];          // 16*8*256*256 = 8388608
  const int images = (int)(N / (256 * 256));     // 128
  dim3 grid(256 / TILE, 256 / TILE, images);     // (4,4,128)
  dim3 block(64, 4, 1);                          // 256 threads = 8 waves
  hipLaunchKernelGGL(cml2d_stats_kernel, grid, block, 0, stream,
                     drive, K_local, out, N);
}